// VectorQuantization_17497696763999
// MI455X (gfx1250) — compile-verified
//
#include <hip/hip_runtime.h>
#include <hip/hip_bf16.h>
#include <float.h>

// ---------------------------------------------------------------------------
// VQ-VAE vector quantization for MI455X (gfx1250).
// Pass 1: argmin_n ( ||e_n||^2 - 2 x.e_n )  via V_WMMA_F32_16X16X4_F32
// Pass 2: gather codebook rows -> quantized, partial sums of (q-x)^2
// Pass 3: deterministic final reduction -> commitment loss
// ---------------------------------------------------------------------------

typedef __attribute__((ext_vector_type(2))) float v2f;
typedef __attribute__((ext_vector_type(8))) float v8f;

#define VQ_D        64          // embedding dim
#define VQ_K        1024        // codebook size
#define CHUNK_COLS  256         // codebook columns staged per LDS chunk
#define LDS_STRIDE  68          // padded row stride (floats) -> conflict-free ds_load_b64
#define ROWS_PER_WAVE 16
#define WAVES_PER_BLOCK 8
#define ROWS_PER_BLOCK (ROWS_PER_WAVE * WAVES_PER_BLOCK)   // 128

// ---------------------------------------------------------------------------
// Kernel 1: per-row argmin over the codebook using f32 WMMA.
//   grid.x = nrows / 128, block = 256 threads (8 wave32)
// ---------------------------------------------------------------------------
__global__ __launch_bounds__(256)
void vq_argmin_kernel(const float* __restrict__ inp,
                      const float* __restrict__ weight,
                      int* __restrict__ ws_idx,
                      float* __restrict__ idx_out,
                      int nrows) {
    __shared__ float wtile[CHUNK_COLS * LDS_STRIDE];  // ~68 KB
    __shared__ float e2s[CHUNK_COLS];

    const int t    = threadIdx.x;
    const int lane = t & 31;
    const int wave = t >> 5;
    const int lo16 = lane & 15;
    const int hi   = lane >> 4;      // 0: k%4 in {0,1}, 1: k%4 in {2,3}

    const int row0 = blockIdx.x * ROWS_PER_BLOCK + wave * ROWS_PER_WAVE;

    // ---- Load this wave's 16x64 A tile into registers (WMMA A layout) ----
    // Lane holds row (lane%16), elements k = 4*s + 2*hi + {0,1}, s = 0..15.
    v2f a[16];
    {
        const float* arow = inp + (size_t)(row0 + lo16) * VQ_D + 2 * hi;
        #pragma unroll
        for (int s = 0; s < 16; ++s)
            a[s] = *(const v2f*)(arow + 4 * s);
    }

    float minv[8];
    int   mini[8];
    #pragma unroll
    for (int r = 0; r < 8; ++r) { minv[r] = FLT_MAX; mini[r] = 0; }

    for (int chunk = 0; chunk < VQ_K / CHUNK_COLS; ++chunk) {
        __syncthreads();
        // ---- Cooperatively stage 256 codebook rows into padded LDS ----
        {
            const float* src = weight + (size_t)(chunk * CHUNK_COLS + t) * VQ_D;
            float* dst = &wtile[t * LDS_STRIDE];
            #pragma unroll
            for (int j = 0; j < VQ_D / 4; ++j)
                *(float4*)(dst + 4 * j) = *(const float4*)(src + 4 * j);
            // per-column squared norm
            float s2 = 0.0f;
            #pragma unroll
            for (int k = 0; k < VQ_D; ++k) { float v = dst[k]; s2 += v * v; }
            e2s[t] = s2;
        }
        __syncthreads();

        // ---- 16 column tiles of 16 codes each ----
        for (int t16 = 0; t16 < 16; ++t16) {
            const int lcol = t16 * 16 + lo16;           // column within chunk
            const float e2 = e2s[lcol];
            const float* brow = &wtile[lcol * LDS_STRIDE + 2 * hi];

            v8f c = {};
            #pragma unroll
            for (int s = 0; s < 16; ++s) {
                v2f b = *(const v2f*)(brow + 4 * s);
                c = __builtin_amdgcn_wmma_f32_16x16x4_f32(
                        /*neg_a=*/false, a[s], /*neg_b=*/false, b,
                        /*c_mod=*/(short)0, c,
                        /*reuse_a=*/false, /*reuse_b=*/false);
            }

            const int ncol = chunk * CHUNK_COLS + lcol; // global code index
            #pragma unroll
            for (int r = 0; r < 8; ++r) {
                // distance (minus constant ||x||^2): ||e||^2 - 2 x.e
                float dist = __builtin_fmaf(-2.0f, c[r], e2);
                if (dist < minv[r]) { minv[r] = dist; mini[r] = ncol; }
            }
        }
    }

    // ---- Cross-lane argmin within each 16-lane half (wave32) ----
    #pragma unroll
    for (int r = 0; r < 8; ++r) {
        float v = minv[r];
        int   i = mini[r];
        #pragma unroll
        for (int off = 8; off >= 1; off >>= 1) {
            float ov = __shfl_xor(v, off, 32);
            int   oi = __shfl_xor(i, off, 32);
            if (ov < v || (ov == v && oi < i)) { v = ov; i = oi; }
        }
        minv[r] = v; mini[r] = i;
    }

    // Register r holds row r (lanes 0-15) / row r+8 (lanes 16-31).
    if (lo16 == 0) {
        const int rbase = row0 + 8 * hi;
        #pragma unroll
        for (int r = 0; r < 8; ++r) {
            ws_idx[rbase + r]  = mini[r];
            idx_out[rbase + r] = (float)mini[r];
        }
    }
}

// ---------------------------------------------------------------------------
// Kernel 2: gather weight[idx] -> quantized, per-block partial loss sums.
//   4 threads per row, 64 rows per block of 256 threads.
// ---------------------------------------------------------------------------
__global__ __launch_bounds__(256)
void vq_gather_kernel(const float* __restrict__ inp,
                      const float* __restrict__ weight,
                      const int* __restrict__ ws_idx,
                      float* __restrict__ quant_out,
                      float* __restrict__ partial) {
    __shared__ float red[256];
    const int t   = threadIdx.x;
    const int row = blockIdx.x * 64 + (t >> 2);
    const int seg = t & 3;                       // 16 floats per segment

    const int e = ws_idx[row];
    const float4* ws = (const float4*)(weight + (size_t)e * VQ_D + seg * 16);
    const float4* xs = (const float4*)(inp + (size_t)row * VQ_D + seg * 16);
    float4*       qs = (float4*)(quant_out + (size_t)row * VQ_D + seg * 16);

    float acc = 0.0f;
    #pragma unroll
    for (int j = 0; j < 4; ++j) {
        float4 wv = ws[j];
        float4 xv = xs[j];
        qs[j] = wv;
        float dx = wv.x - xv.x; acc += dx * dx;
        float dy = wv.y - xv.y; acc += dy * dy;
        float dz = wv.z - xv.z; acc += dz * dz;
        float dw = wv.w - xv.w; acc += dw * dw;
    }

    red[t] = acc;
    __syncthreads();
    #pragma unroll
    for (int s = 128; s > 0; s >>= 1) {
        if (t < s) red[t] += red[t + s];
        __syncthreads();
    }
    if (t == 0) partial[blockIdx.x] = red[0];
}

// ---------------------------------------------------------------------------
// Kernel 3: fixed-order final reduction of partial sums -> loss scalar.
// ---------------------------------------------------------------------------
__global__ __launch_bounds__(256)
void vq_loss_kernel(const float* __restrict__ partial,
                    float* __restrict__ loss_out,
                    int nparts, float inv_count) {
    __shared__ float red[256];
    const int t = threadIdx.x;
    float acc = 0.0f;
    for (int i = t; i < nparts; i += 256) acc += partial[i];
    red[t] = acc;
    __syncthreads();
    #pragma unroll
    for (int s = 128; s > 0; s >>= 1) {
        if (t < s) red[t] += red[t + s];
        __syncthreads();
    }
    // commitment = COST*(q_loss + e_loss) = 0.25 * 2 * mean = 0.5 * mean
    if (t == 0) *loss_out = 0.5f * red[0] * inv_count;
}

// ---------------------------------------------------------------------------
extern "C" void kernel_launch(void* const* d_in, const int* in_sizes, int n_in,
                              void* d_out, int out_size, void* d_ws, size_t ws_size,
                              hipStream_t stream) {
    const float* inp    = (const float*)d_in[0];   // (32,64,32,32) f32
    const float* weight = (const float*)d_in[1];   // (1024,64) f32

    const int nelem = in_sizes[0];                 // 2,097,152
    const int nrows = nelem / VQ_D;                // 32,768

    float* out       = (float*)d_out;
    float* quant_out = out;                        // [nelem]
    float* idx_out   = out + nelem;                // [nrows] (indices as f32)
    float* loss_out  = out + nelem + nrows;        // [1]

    int*   ws_idx     = (int*)d_ws;                // [nrows]
    float* ws_partial = (float*)d_ws + nrows;      // [nrows/64]

    const int gather_blocks = nrows / 64;          // 512

    vq_argmin_kernel<<<nrows / ROWS_PER_BLOCK, 256, 0, stream>>>(
        inp, weight, ws_idx, idx_out, nrows);

    vq_gather_kernel<<<gather_blocks, 256, 0, stream>>>(
        inp, weight, ws_idx, quant_out, ws_partial);

    vq_loss_kernel<<<1, 256, 0, stream>>>(
        ws_partial, loss_out, gather_blocks, 1.0f / (float)nelem);
}